// CnnBasedRnn_73770358276975
// MI455X (gfx1250) — compile-verified
//
#include <hip/hip_runtime.h>

// ---------------------------------------------------------------------------
// CnnBasedRnn for MI455X (gfx1250).  B=16, L=256, D=64, 2 layers.
//  Pass A (throughput): x-only conv contribution (+bias) for output rows 0..32
//    as a banded GEMM on V_WMMA_F32_16X16X4_F32. Zero-padded operands give a
//    compile-time trip count: 3x6 fully-unrolled K-slices, straight-line
//    ds_load + v_wmma, no EXEC games (WMMA needs EXEC all-1s).
//  Pass B (latency): sequential scan, one workgroup per batch image. h kept in
//    LDS (ping-pong, zero halos). Next step's xc slab is double-buffered into
//    LDS with global_load_async_to_lds_b128 + s_wait_asynccnt.
// Workspace: xc buffer = 16*256*33*64 floats = 34.6 MB in d_ws.
// ---------------------------------------------------------------------------

typedef float v2f __attribute__((ext_vector_type(2)));
typedef float v8f __attribute__((ext_vector_type(8)));
typedef int   v4i __attribute__((vector_size(16)));   // matches builtin param

#define BB 16
#define LL 256
#define DD 64
#define XCROWS 33
#define XSC (DD + 10)        // 74 cols: 4 zero + 64 data + 4 zero + 2 spare
#define TROWS (DD + 8)       // 72 rows: j = -4 .. 67 (storage = j+4)

#define HAVE_WMMA_F32X4 __has_builtin(__builtin_amdgcn_wmma_f32_16x16x4_f32)

#if __has_builtin(__builtin_amdgcn_global_load_async_to_lds_b128) && \
    __has_builtin(__builtin_amdgcn_s_wait_asynccnt)
#define ASYNC_OK 1
#else
#define ASYNC_OK 0
#endif

#if ASYNC_OK
__device__ __forceinline__ void async_b128(const void* g, void* l) {
  __builtin_amdgcn_global_load_async_to_lds_b128(
      (__attribute__((address_space(1))) v4i*)(uintptr_t)g,
      (__attribute__((address_space(3))) v4i*)(uintptr_t)l, 0, 0);
}
#endif

// ============================ Pass A =======================================
// out[r][c] = bias + sum_kr sum_j X_kr[r][j] * T_kr[j][c]
//   X_kr[r][j] = in[2r-1+kr][j]   (row -1 and cols outside [0,64) zero-padded)
//   T_kr[j][c] = W[kr][j-c+1] on the tridiagonal band, else 0 (j padded too).
// One block per (b,t) image: 8 waves = 2 row-tiles x 4 col-tiles of 16x16.
__global__ __launch_bounds__(256) void xcontrib_kernel(
    const float* __restrict__ src,   // (B,L,64,64) layer input
    float* __restrict__ xc,          // (B,L,33,64) output
    const float* __restrict__ Wl,    // 9 weights for this layer
    const float* __restrict__ bl)    // 1 bias
{
  __shared__ float xs[DD + 1][XSC];        // row 0 = zeros; col storage = c+4
  __shared__ float Tm[3][TROWS][DD + 2];   // banded T; row storage = j+4

  const int img = blockIdx.x;              // b*L + t
  const float* sp = src + (size_t)img * DD * DD;

  const float w00 = Wl[0], w01 = Wl[1], w02 = Wl[2];
  const float w10 = Wl[3], w11 = Wl[4], w12 = Wl[5];
  const float w20 = Wl[6], w21 = Wl[7], w22 = Wl[8];
  const float bias = bl[0];

  // Zero halos: pad row 0, and side columns 0..3 / 68..71 of data rows.
  if (threadIdx.x < XSC) xs[0][threadIdx.x] = 0.f;
  for (int i = threadIdx.x; i < DD * 8; i += 256) {
    const int r = i >> 3, k = i & 7;
    xs[1 + r][(k < 4) ? k : (DD + k)] = 0.f;   // cols 0..3, 68..71
  }
  // Stage the 16 KB image (coalesced), shifted by the 4-col halo.
  for (int i = threadIdx.x; i < DD * DD; i += 256)
    xs[1 + (i >> 6)][4 + (i & 63)] = sp[i];

  // Build T with padded j rows; out-of-band rows must be real zeros
  // (0 x NaN = NaN in WMMA, so no uninitialized LDS may be read).
  for (int e = threadIdx.x; e < TROWS * DD; e += 256) {
    const int jb = e >> 6, cc = e & 63;
    const int j = jb - 4;
    const int d = ((unsigned)j < (unsigned)DD) ? (j - cc + 1) : 99;
    Tm[0][jb][cc] = (d == 0) ? w00 : (d == 1) ? w01 : (d == 2) ? w02 : 0.f;
    Tm[1][jb][cc] = (d == 0) ? w10 : (d == 1) ? w11 : (d == 2) ? w12 : 0.f;
    Tm[2][jb][cc] = (d == 0) ? w20 : (d == 1) ? w21 : (d == 2) ? w22 : 0.f;
  }
  __syncthreads();

  const int lane = threadIdx.x & 31;
  const int wv   = threadIdx.x >> 5;     // wave 0..7
  const int r0   = (wv >> 2) * 16;       // output row tile: 0 or 16
  const int n0   = (wv & 3) * 16;        // output col tile: 0,16,32,48
  const int m    = lane & 15;
  const int hf   = lane >> 4;            // half-wave: lanes>=16 hold K+2..K+3
  const int n    = n0 + m;

  float* op = xc + (size_t)img * XCROWS * DD;

#if HAVE_WMMA_F32X4
  v8f acc;
#pragma unroll
  for (int v = 0; v < 8; ++v) acc[v] = bias;   // fold bias into C

  // Fixed 6 K-slices starting at j = n0-4 (storage jb = n0): covers the whole
  // band [n0-1, n0+16]; out-of-band reads hit zero padding. Fully unrolled.
#pragma unroll
  for (int kr = 0; kr < 3; ++kr) {
    // A rows: lane m -> output row r0+m -> input row 2(r0+m)-1+kr -> xs row +1.
    const float* arow = &xs[2 * (r0 + m) + kr][0];
#pragma unroll
    for (int s = 0; s < 6; ++s) {
      const int jb = n0 + 4 * s + 2 * hf;      // storage index (= j+4)
      v2f a, bm;
      a.x  = arow[jb];
      a.y  = arow[jb + 1];
      bm.x = Tm[kr][jb][n];                    // B 4x16: lane = column n
      bm.y = Tm[kr][jb + 1][n];
      acc = __builtin_amdgcn_wmma_f32_16x16x4_f32(
          false, a, false, bm, (short)0, acc, false, false);
    }
  }

  // C/D layout: lane<16 -> (M=v, N=lane); lane>=16 -> (M=v+8, N=lane-16).
  const int orow0 = r0 + (hf ? 8 : 0);
#pragma unroll
  for (int v = 0; v < 8; ++v)
    op[(size_t)(orow0 + v) * DD + n] = acc[v];
#else
  // VALU fallback (build safety): rows 0..31 direct 9-tap, halo-padded.
  for (int idx = threadIdx.x; idx < 32 * DD; idx += 256) {
    const int r = idx >> 6, c = idx & 63;
    float s = bias;
    const float wrow[3][3] = {{w00,w01,w02},{w10,w11,w12},{w20,w21,w22}};
#pragma unroll
    for (int kr = 0; kr < 3; ++kr) {
      const float* arow = &xs[2 * r + kr][0];
      s += wrow[kr][0] * arow[c + 3] + wrow[kr][1] * arow[c + 4]
         + wrow[kr][2] * arow[c + 5];
    }
    op[(size_t)r * DD + c] = s;
  }
#endif

  // Row 32: only the kr=0 tap reads x (input row 63 = xs[64]); halo-padded.
  if (threadIdx.x < DD) {
    const int c = threadIdx.x;
    op[32 * DD + c] = bias + w00 * xs[64][c + 3] + w01 * xs[64][c + 4]
                           + w02 * xs[64][c + 5];
  }
}

// ============================ Pass B =======================================
// One workgroup per batch image; 256 threads; 256 sequential steps.
// h ping-pong in LDS: [2][64][66], columns 0 and 65 are zero halos.
// xc slab for step t+1 is async-loaded to LDS while step t computes.
__global__ __launch_bounds__(256) void scan_kernel(
    const float* __restrict__ xc,    // (B,L,33,64)
    float* __restrict__ out,         // (B,L,64,64)
    const float* __restrict__ Wl,
    const float* __restrict__ bl)
{
  __shared__ float h[2][DD][DD + 2];
#if ASYNC_OK
  __shared__ __align__(16) float stage[2][XCROWS * DD];   // 2 x 8.25 KB
#endif

  const int b  = blockIdx.x;
  const int c  = threadIdx.x & 63;
  const int rb = threadIdx.x >> 6;   // row block 0..3 (16 rows each)

  const float w00 = Wl[0], w01 = Wl[1], w02 = Wl[2];
  const float w10 = Wl[3], w11 = Wl[4], w12 = Wl[5];
  const float w20 = Wl[6], w21 = Wl[7], w22 = Wl[8];
  const float bias = bl[0];

  const float* xcb = xc + (size_t)b * LL * XCROWS * DD;
  float* ob = out + (size_t)b * LL * DD * DD;

#if ASYNC_OK
  // Prologue: async-stage xc(t=0) while we zero h.
  for (int ch = threadIdx.x; ch < (XCROWS * DD) / 4; ch += 256)
    async_b128(xcb + ch * 4, &stage[0][ch * 4]);
#endif
  for (int i = threadIdx.x; i < 2 * DD * (DD + 2); i += 256)
    (&h[0][0][0])[i] = 0.f;
#if ASYNC_OK
  __builtin_amdgcn_s_wait_asynccnt(0);
#endif
  __syncthreads();

  for (int t = 0; t < LL; ++t) {
    const int pr = t & 1;            // h_{t-1} and xc(t) live here
    const int nx = pr ^ 1;
    float* ot = ob + (size_t)t * DD * DD;

#if ASYNC_OK
    if (t + 1 < LL) {                // overlap next slab's fetch with compute
      const float* g = xcb + (size_t)(t + 1) * XCROWS * DD;
      for (int ch = threadIdx.x; ch < (XCROWS * DD) / 4; ch += 256)
        async_b128(g + ch * 4, &stage[nx][ch * 4]);
    }
    const float* xrow = &stage[pr][0];
#else
    const float* xrow = xcb + (size_t)t * XCROWS * DD;
    if (t + 1 < LL)
      __builtin_prefetch(xcb + (size_t)(t + 1) * XCROWS * DD + rb * 8 * DD + c, 0, 0);
#endif

#pragma unroll
    for (int i = 0; i < 16; ++i) {
      const int r = rb * 16 + i;
      float v;
      if (r < 32) {                  // pure x rows
        v = xrow[r * DD + c];
      } else if (r == 32) {          // + h rows 0,1 (kr=1,2)
        v = xrow[32 * DD + c];
        v += w10 * h[pr][0][c] + w11 * h[pr][0][c + 1] + w12 * h[pr][0][c + 2];
        v += w20 * h[pr][1][c] + w21 * h[pr][1][c + 1] + w22 * h[pr][1][c + 2];
      } else {                       // pure h rows: base = 2r-65 in [1,61]
        const int base = 2 * r - 65;
        v = bias;
        v += w00 * h[pr][base    ][c] + w01 * h[pr][base    ][c + 1] + w02 * h[pr][base    ][c + 2];
        v += w10 * h[pr][base + 1][c] + w11 * h[pr][base + 1][c + 1] + w12 * h[pr][base + 1][c + 2];
        v += w20 * h[pr][base + 2][c] + w21 * h[pr][base + 2][c + 1] + w22 * h[pr][base + 2][c + 2];
      }
      // tanh(v) = 1 - 2/(e^{2v}+1); exact saturation, one v_exp_f32.
      const float e = __expf(2.f * v);
      const float o = 1.f - 2.f / (e + 1.f);
      ot[r * DD + c] = o;
      h[nx][r][c + 1] = o;           // interior cols 1..64; halos stay 0
    }

#if ASYNC_OK
    __builtin_amdgcn_s_wait_asynccnt(0);   // stage[nx] resident before t+1
#endif
    __syncthreads();                 // single barrier per step (ping-pong)
  }
}

// ============================ Launch =======================================
extern "C" void kernel_launch(void* const* d_in, const int* in_sizes, int n_in,
                              void* d_out, int out_size, void* d_ws, size_t ws_size,
                              hipStream_t stream) {
  const float* x  = (const float*)d_in[0];   // (16,256,64,64)
  const float* W  = (const float*)d_in[1];   // (2,1,1,3,3)
  const float* bv = (const float*)d_in[2];   // (2,)
  float* out = (float*)d_out;
  float* xc  = (float*)d_ws;                 // needs 16*256*33*64*4 = 34.6 MB

  const dim3 blk(256), gA(BB * LL), gS(BB);

  // Layer 0
  xcontrib_kernel<<<gA, blk, 0, stream>>>(x,   xc, W + 0, bv + 0);
  scan_kernel   <<<gS, blk, 0, stream>>>(xc, out, W + 0, bv + 0);
  // Layer 1 (reads layer-0 output from d_out, then overwrites it)
  xcontrib_kernel<<<gA, blk, 0, stream>>>(out, xc, W + 9, bv + 1);
  scan_kernel   <<<gS, blk, 0, stream>>>(xc, out, W + 9, bv + 1);
}